// DynamicTimeStretch_38543036514467
// MI455X (gfx1250) — compile-verified
//
#include <hip/hip_runtime.h>
#include <math.h>

// Problem constants (must match the reference file)
#define NBATCH   16
#define N_FREQ   513
#define T_IN     2048
#define N_OUT    2276      // ceil(2048 / 0.9)
#define RATE     0.9f
#define NTHREADS 256

#ifndef __has_builtin
#define __has_builtin(x) 0
#endif

// The gfx1250 async-LDS builtins take int4-vector pointers (b128 payload):
//   __builtin_amdgcn_global_load_async_to_lds_b128(v4i AS1* src, v4i AS3* dst,
//                                                  imm offset, imm cpol)
typedef int v4i __attribute__((vector_size(16)));
typedef __attribute__((address_space(1))) v4i as1_v4i;   // global
typedef __attribute__((address_space(3))) v4i as3_v4i;   // LDS

#if __has_builtin(__builtin_amdgcn_global_load_async_to_lds_b128)
#define HAVE_ASYNC_LDS 1
#else
#define HAVE_ASYNC_LDS 0
#endif

__device__ __forceinline__ void wait_asynccnt0() {
#if __has_builtin(__builtin_amdgcn_s_wait_asynccnt)
    __builtin_amdgcn_s_wait_asynccnt(0);
#else
    asm volatile("s_wait_asynccnt 0" ::: "memory");
#endif
}

// One workgroup per (b, freq) row.
// Stage the full 2048-frame complex row into LDS (async global->LDS DMA path,
// ASYNCcnt-tracked), then each thread produces 9 strided output frames via
// fractional-index magnitude interpolation.  The reference's phase
// accumulation is dead code for |.|^2, so it is omitted entirely.
__global__ __launch_bounds__(NTHREADS) void pv_power_kernel(
    const float* __restrict__ xr,
    const float* __restrict__ xi,
    float* __restrict__ out)
{
    __shared__ __align__(16) float lds_re[T_IN + 4];
    __shared__ __align__(16) float lds_im[T_IN + 4];

    const int row = blockIdx.x;            // b * N_FREQ + f
    const int tid = threadIdx.x;

    const float* gre = xr + (size_t)row * T_IN;
    const float* gim = xi + (size_t)row * T_IN;

    // Zero-pad frame T_IN (reference pads spec with 2 zero frames; only
    // idx+1 == T_IN is ever touched since max idx = 2047).
    if (tid < 4) {
        lds_re[T_IN + tid] = 0.0f;
        lds_im[T_IN + tid] = 0.0f;
    }

#if HAVE_ASYNC_LDS
    {
        // 2048 floats = 512 x b128 per array; 256 threads x 2 iters each.
        const int o0 = tid * 4;
        const int o1 = (tid + NTHREADS) * 4;
        __builtin_amdgcn_global_load_async_to_lds_b128(
            (as1_v4i*)(gre + o0), (as3_v4i*)(lds_re + o0), 0, 0);
        __builtin_amdgcn_global_load_async_to_lds_b128(
            (as1_v4i*)(gre + o1), (as3_v4i*)(lds_re + o1), 0, 0);
        __builtin_amdgcn_global_load_async_to_lds_b128(
            (as1_v4i*)(gim + o0), (as3_v4i*)(lds_im + o0), 0, 0);
        __builtin_amdgcn_global_load_async_to_lds_b128(
            (as1_v4i*)(gim + o1), (as3_v4i*)(lds_im + o1), 0, 0);
        wait_asynccnt0();      // this wave's async transfers landed in LDS
    }
#else
    {
        // Fallback: vectorized b128 loads through VGPRs into LDS.
        const float4* gre4 = (const float4*)gre;
        const float4* gim4 = (const float4*)gim;
        float4* lre4 = (float4*)lds_re;
        float4* lim4 = (float4*)lds_im;
#pragma unroll
        for (int k = 0; k < 2; ++k) {
            const int i = tid + k * NTHREADS;
            lre4[i] = gre4[i];
            lim4[i] = gim4[i];
        }
    }
#endif
    __syncthreads();           // make all waves' LDS writes visible

    float* orow = out + (size_t)row * N_OUT;
#pragma unroll
    for (int k = 0; k < (N_OUT + NTHREADS - 1) / NTHREADS; ++k) {
        const int t = tid + k * NTHREADS;
        if (t < N_OUT) {
            // Match jnp float32 arithmetic: pos = f32(t) * f32(0.9),
            // idx = trunc(pos), alpha = mod(pos, 1) = pos - trunc(pos).
            const float pos   = (float)t * RATE;
            const int   it    = (int)pos;
            const float alpha = pos - (float)it;

            const float re0 = lds_re[it];
            const float im0 = lds_im[it];
            const float re1 = lds_re[it + 1];
            const float im1 = lds_im[it + 1];

            const float n0 = sqrtf(fmaf(re0, re0, im0 * im0));
            const float n1 = sqrtf(fmaf(re1, re1, im1 * im1));
            const float m  = fmaf(alpha, n1, (1.0f - alpha) * n0);
            orow[t] = m * m;   // |mag * e^{i phase}|^2 == mag^2
        }
    }
}

extern "C" void kernel_launch(void* const* d_in, const int* in_sizes, int n_in,
                              void* d_out, int out_size, void* d_ws, size_t ws_size,
                              hipStream_t stream) {
    (void)in_sizes; (void)n_in; (void)d_ws; (void)ws_size; (void)out_size;
    const float* xr  = (const float*)d_in[0];
    const float* xi  = (const float*)d_in[1];
    float*       out = (float*)d_out;

    const dim3 grid(NBATCH * N_FREQ);   // 8208 rows
    const dim3 block(NTHREADS);
    hipLaunchKernelGGL(pv_power_kernel, grid, block, 0, stream, xr, xi, out);
}